// FeatureFusionBlock_75479755260663
// MI455X (gfx1250) — compile-verified
//
#include <hip/hip_runtime.h>

#define HD 512
#define WD 512
// ---- attention kernel tiling: 32 wide x 16 tall, 2 vertical px / thread ----
#define ATX 32
#define ATY 16
#define AHW (ATX + 2)               // 34
#define AHH (ATY + 2)               // 18
#define APLANE (AHH * AHW)          // 612
#define ANLDS  (2 * APLANE)         // 1224 floats
// ---- conv kernel tiling: 32 x 8, 1 px / thread ----
#define CTX 32
#define CTY 8
#define CHW (CTX + 2)               // 34
#define CHH (CTY + 2)               // 10
#define CPLANE_T (CHH * CHW)        // 340
#define CNLDS  (2 * CPLANE_T)       // 680
#define NPIX    ((size_t)HD * WD)   // 262144
#define CPLANE  (2 * NPIX)          // per-batch 2-channel plane

typedef float v2f __attribute__((ext_vector_type(2)));

__device__ __forceinline__ void async_lds_b32(unsigned ldsOff, const void* src) {
    asm volatile("global_load_async_to_lds_b32 %0, %1, off"
                 :: "v"(ldsOff), "v"((unsigned long long)src)
                 : "memory");
}
__device__ __forceinline__ void wait_async0() {
    asm volatile("s_wait_asynccnt 0" ::: "memory");
}

// ---------------------------------------------------------------------------
// Fused relu -> attn_conv (+ up to two residual adds). 2 pixels / thread.
// Tile staged raw into LDS via CDNA5 async-LDS DMA, then one RMW pass applies
// relu + zero-padding so all 9-neighbor consumers read clean values.
// ---------------------------------------------------------------------------
__global__ __launch_bounds__(256) void attn_kernel(
    const float* __restrict__ x,
    const float* __restrict__ res1,   // nullable
    const float* __restrict__ res2,   // nullable
    float* __restrict__ out,
    const float* __restrict__ wq, const float* __restrict__ bq,
    const float* __restrict__ wk, const float* __restrict__ bk,
    const float* __restrict__ wv, const float* __restrict__ bv,
    const float* __restrict__ rh, const float* __restrict__ rw,
    int aidx)
{
    __shared__ float tile[ANLDS];     // [c][yy][xx]

    const int tid = threadIdx.y * ATX + threadIdx.x;
    const int x0  = blockIdx.x * ATX;
    const int y0  = blockIdx.y * ATY;
    const int b   = blockIdx.z;
    const size_t base = (size_t)b * CPLANE;

    // ---- async DMA of raw tile into LDS (clamped addresses) ----
    for (int t = tid; t < ANLDS; t += 256) {
        int c  = t / APLANE;
        int r  = t % APLANE;
        int yy = r / AHW;
        int xx = r % AHW;
        int cy = min(max(y0 + yy - 1, 0), HD - 1);
        int cx = min(max(x0 + xx - 1, 0), WD - 1);
        async_lds_b32((unsigned)(t * 4), x + base + ((size_t)c * HD + cy) * WD + cx);
    }
    wait_async0();

    // one pass: relu in-image values, zero out-of-image halo (pad semantics)
    for (int t = tid; t < ANLDS; t += 256) {
        int r  = t % APLANE;
        int yy = r / AHW;
        int xx = r % AHW;
        int gy = y0 + yy - 1;
        int gx = x0 + xx - 1;
        bool inb = (gy >= 0) & (gy < HD) & (gx >= 0) & (gx < WD);
        tile[t] = inb ? fmaxf(tile[t], 0.0f) : 0.0f;
    }
    __syncthreads();

    // ---- per-layer weights (uniform -> scalar regs) ----
    const float wq00 = wq[aidx*4+0], wq01 = wq[aidx*4+1];
    const float wq10 = wq[aidx*4+2], wq11 = wq[aidx*4+3];
    const float bq0  = bq[aidx*2+0], bq1  = bq[aidx*2+1];
    const float wk00 = wk[aidx*4+0], wk01 = wk[aidx*4+1];
    const float wk10 = wk[aidx*4+2], wk11 = wk[aidx*4+3];
    const float bk0  = bk[aidx*2+0], bk1  = bk[aidx*2+1];
    const float wv00 = wv[aidx*4+0], wv01 = wv[aidx*4+1];
    const float wv10 = wv[aidx*4+2], wv11 = wv[aidx*4+3];
    const float bv0  = bv[aidx*2+0], bv1  = bv[aidx*2+1];
    // scalar-folded bias + positional offsets
    const float C0[3] = { bk0 + rh[aidx*3+0], bk0 + rh[aidx*3+1], bk0 + rh[aidx*3+2] };
    const float C1[3] = { bk1 + rw[aidx*3+0], bk1 + rw[aidx*3+1], bk1 + rw[aidx*3+2] };

    const int tx = threadIdx.x, ty = threadIdx.y;
    const int ry0 = 2 * ty;                   // first output row in tile
    const float* t0 = &tile[0];
    const float* t1 = &tile[APLANE];

    // 12 shared neighbor locations (4 rows x 3 cols), k-base and v per channel
    float kb0[4][3], kb1[4][3], vv0[4][3], vv1[4][3];
    float cen0[2], cen1[2];
#pragma unroll
    for (int r = 0; r < 4; ++r) {
#pragma unroll
        for (int cx = 0; cx < 3; ++cx) {
            const float a0 = t0[(ry0 + r) * AHW + tx + cx];
            const float a1 = t1[(ry0 + r) * AHW + tx + cx];
            kb0[r][cx] = wk00 * a0 + wk01 * a1;           // + C0[di] later
            kb1[r][cx] = wk10 * a0 + wk11 * a1;           // + C1[dj] later
            vv0[r][cx] = wv00 * a0 + (wv01 * a1 + bv0);
            vv1[r][cx] = wv10 * a0 + (wv11 * a1 + bv1);
            if (cx == 1 && (r == 1 || r == 2)) { cen0[r-1] = a0; cen1[r-1] = a1; }
        }
    }

#pragma unroll
    for (int p = 0; p < 2; ++p) {
        const float q0 = wq00 * cen0[p] + (wq01 * cen1[p] + bq0);
        const float q1 = wq10 * cen0[p] + (wq11 * cen1[p] + bq1);

        float l0[9], l1[9];
        float m0 = -3.0e38f, m1 = -3.0e38f;
#pragma unroll
        for (int di = 0; di < 3; ++di) {
#pragma unroll
            for (int dj = 0; dj < 3; ++dj) {
                const int j = di * 3 + dj;
                l0[j] = q0 * (kb0[p + di][dj] + C0[di]);
                l1[j] = q1 * (kb1[p + di][dj] + C1[dj]);
                m0 = fmaxf(m0, l0[j]);
                m1 = fmaxf(m1, l1[j]);
            }
        }
        float s0 = 0.f, s1 = 0.f, acc0 = 0.f, acc1 = 0.f;
#pragma unroll
        for (int di = 0; di < 3; ++di) {
#pragma unroll
            for (int dj = 0; dj < 3; ++dj) {
                const int j = di * 3 + dj;
                const float e0 = __expf(l0[j] - m0);
                const float e1 = __expf(l1[j] - m1);
                s0 += e0; s1 += e1;
                acc0 += e0 * vv0[p + di][dj];
                acc1 += e1 * vv1[p + di][dj];
            }
        }
        float o0 = acc0 / s0;
        float o1 = acc1 / s1;

        const size_t idx0 = base + (size_t)(y0 + ry0 + p) * WD + (x0 + tx);
        const size_t idx1 = idx0 + NPIX;
        if (res1) { o0 += res1[idx0]; o1 += res1[idx1]; }
        if (res2) { o0 += res2[idx0]; o1 += res2[idx1]; }
        out[idx0] = o0;
        out[idx1] = o1;
    }
}

// ---------------------------------------------------------------------------
// reflect-pad 3x3 conv (2 -> 1 ch) + sigmoid, async-LDS staged tile
// (reflection folded into the DMA source addresses -> no fixup pass)
// ---------------------------------------------------------------------------
__global__ __launch_bounds__(256) void conv_sig_kernel(
    const float* __restrict__ in,
    const float* __restrict__ cw, const float* __restrict__ cb,
    float* __restrict__ out)
{
    __shared__ float tile[CNLDS];     // [c][yy][xx]

    const int tid = threadIdx.y * CTX + threadIdx.x;
    const int x0  = blockIdx.x * CTX;
    const int y0  = blockIdx.y * CTY;
    const int b   = blockIdx.z;
    const size_t base = (size_t)b * CPLANE;

    for (int t = tid; t < CNLDS; t += 256) {
        int c  = t / CPLANE_T;
        int r  = t % CPLANE_T;
        int yy = r / CHW;
        int xx = r % CHW;
        int gy = y0 + yy - 1;
        int gx = x0 + xx - 1;
        int sy = (gy < 0) ? -gy : ((gy >= HD) ? 2 * HD - 2 - gy : gy);  // reflect
        int sx = (gx < 0) ? -gx : ((gx >= WD) ? 2 * WD - 2 - gx : gx);
        async_lds_b32((unsigned)(t * 4), in + base + ((size_t)c * HD + sy) * WD + sx);
    }
    wait_async0();
    __syncthreads();

    const float w0[9] = { cw[0],cw[1],cw[2],cw[3],cw[4],cw[5],cw[6],cw[7],cw[8] };
    const float w1[9] = { cw[9],cw[10],cw[11],cw[12],cw[13],cw[14],cw[15],cw[16],cw[17] };

    const int tx = threadIdx.x, ty = threadIdx.y;
    const float* t0 = &tile[0];
    const float* t1 = &tile[CPLANE_T];

    float acc = cb[0];
#pragma unroll
    for (int ky = 0; ky < 3; ++ky) {
#pragma unroll
        for (int kx = 0; kx < 3; ++kx) {
            acc += w0[ky*3+kx] * t0[(ty + ky) * CHW + tx + kx];
            acc += w1[ky*3+kx] * t1[(ty + ky) * CHW + tx + kx];
        }
    }
    out[(size_t)b * NPIX + (size_t)(y0 + ty) * WD + (x0 + tx)] =
        1.0f / (1.0f + __expf(-acc));
}

// ---------------------------------------------------------------------------
// 2x bilinear upsample (linspace(0,H-1,2H) weights), NT b64 stores
// ---------------------------------------------------------------------------
__global__ __launch_bounds__(256) void up_kernel(
    const float* __restrict__ s, float* __restrict__ d)
{
    const int tg = blockIdx.x * 256 + threadIdx.x;   // per-batch, 2 px/thread
    const int b  = blockIdx.y;
    const int Y  = tg >> 9;          // 0..1023
    const int X0 = (tg & 511) * 2;   // even columns
    const float sc = 511.0f / 1023.0f;

    const float ysf = (float)Y * sc;
    const int   yi0 = (int)ysf;
    const float wy  = ysf - (float)yi0;
    const int   yi1 = min(yi0 + 1, HD - 1);
    const float* r0 = s + ((size_t)b * HD + yi0) * WD;
    const float* r1 = s + ((size_t)b * HD + yi1) * WD;

    v2f o;
#pragma unroll
    for (int k = 0; k < 2; ++k) {
        const float xsf = (float)(X0 + k) * sc;
        const int   xi0 = (int)xsf;
        const float wx  = xsf - (float)xi0;
        const int   xi1 = min(xi0 + 1, WD - 1);
        const float v00 = r0[xi0], v01 = r0[xi1];
        const float v10 = r1[xi0], v11 = r1[xi1];
        o[k] = v00 * (1.f - wy) * (1.f - wx) + v01 * (1.f - wy) * wx
             + v10 * wy * (1.f - wx)        + v11 * wy * wx;
    }
    v2f* dst = (v2f*)(d + ((size_t)b * 1024 + Y) * 1024 + X0);
    __builtin_nontemporal_store(o, dst);
}

// ---------------------------------------------------------------------------
extern "C" void kernel_launch(void* const* d_in, const int* in_sizes, int n_in,
                              void* d_out, int out_size, void* d_ws, size_t ws_size,
                              hipStream_t stream) {
    (void)in_sizes; (void)n_in; (void)out_size; (void)ws_size;
    const float* in1 = (const float*)d_in[0];
    const float* in2 = (const float*)d_in[1];
    const float* wq  = (const float*)d_in[2];
    const float* bq  = (const float*)d_in[3];
    const float* wk  = (const float*)d_in[4];
    const float* bk  = (const float*)d_in[5];
    const float* wv  = (const float*)d_in[6];
    const float* bv  = (const float*)d_in[7];
    const float* rh  = (const float*)d_in[8];
    const float* rw  = (const float*)d_in[9];
    const float* cw  = (const float*)d_in[10];
    const float* cb  = (const float*)d_in[11];

    float* out   = (float*)d_out;
    const size_t SMALL = (size_t)8 * NPIX;       // 2,097,152 (out tuple elem 0)
    const size_t BUF   = (size_t)8 * CPLANE;     // 4,194,304 floats / buffer
    float* A  = (float*)d_ws;                    // 16.8 MB scratch
    float* Bb = out + SMALL;                     // scratch in d_out tail...
    float* Cc = out + SMALL + BUF;               // ...overwritten by up_kernel

    dim3 ablk(ATX, 8, 1);
    dim3 agrd(WD / ATX, HD / ATY, 8);            // (16, 32, 8)
    dim3 cblk(CTX, CTY, 1);
    dim3 cgrd(WD / CTX, HD / CTY, 8);            // (16, 64, 8)

    // rau(input_1, u=0)
    attn_kernel<<<agrd, ablk, 0, stream>>>(in1, nullptr, nullptr, A,  wq,bq,wk,bk,wv,bv,rh,rw, 0);
    attn_kernel<<<agrd, ablk, 0, stream>>>(A,   in1,     nullptr, Bb, wq,bq,wk,bk,wv,bv,rh,rw, 1);
    // rau(input_2, u=1), fused with "+ rau(input_1)"
    attn_kernel<<<agrd, ablk, 0, stream>>>(in2, nullptr, nullptr, A,  wq,bq,wk,bk,wv,bv,rh,rw, 2);
    attn_kernel<<<agrd, ablk, 0, stream>>>(A,   in2,     Bb,      Cc, wq,bq,wk,bk,wv,bv,rh,rw, 3);
    // rau(out, u=2)
    attn_kernel<<<agrd, ablk, 0, stream>>>(Cc,  nullptr, nullptr, A,  wq,bq,wk,bk,wv,bv,rh,rw, 4);
    attn_kernel<<<agrd, ablk, 0, stream>>>(A,   Cc,      nullptr, Bb, wq,bq,wk,bk,wv,bv,rh,rw, 5);
    // reflect-pad conv + sigmoid -> d_out[0 : 2M)
    conv_sig_kernel<<<cgrd, cblk, 0, stream>>>(Bb, cw, cb, out);
    // bilinear x2 -> d_out[2M : 10.5M)  (clobbers Bb/Cc, both dead)
    up_kernel<<<dim3(2048, 8, 1), dim3(256, 1, 1), 0, stream>>>(out, out + SMALL);
}